// CapLayer_26731876450734
// MI455X (gfx1250) — compile-verified
//
#include <hip/hip_runtime.h>
#include <math.h>

// Capsule layer for MI455X (gfx1250, wave32).
// Phase 1: grouped 1x1 conv as per-(b,g) GEMM via V_WMMA_F32_16X16X4_F32,
//          writing pred in [b][j][i][d] layout (the reference's transpose is
//          absorbed into the C-tile -> store mapping).
// Phase 2: 3 rounds of dynamic routing, one block per batch element,
//          logits resident in LDS (46KB), softmax via v_exp_f32,
//          block reductions via wave32 shuffles + LDS float atomics.

typedef __attribute__((ext_vector_type(2))) float v2f;
typedef __attribute__((ext_vector_type(8))) float v8f;

#define G_ 32
#define J_ 10
#define D_ 16
#define I_ 1152      // G * h * w = 32*36
#define HW_ 36
#define DIN_ 8
#define BS_ 256

// ---------------- Phase 1: pred = grouped conv (WMMA f32 16x16x4) ----------
// One wave per (brel, g). M = 160 outputs = 10 tiles (tile == j, row == d),
// N = 36 p-values = 3 tiles of 16 (padded), K = 8 = two k-steps of 4.
__global__ __launch_bounds__(32) void cap_pred_wmma(
    const float* __restrict__ x,     // (256, 256, 36)
    const float* __restrict__ W,     // (5120, 8)
    const float* __restrict__ bias,  // (5120,)
    float* __restrict__ pred,        // (chunk, J, I, D)
    int b0)
{
    const int bg   = blockIdx.x;
    const int brel = bg / G_;
    const int g    = bg % G_;
    const int b    = b0 + brel;

    const int lane = threadIdx.x & 31;
    const int half = lane >> 4;       // 0: lanes 0-15, 1: lanes 16-31
    const int l16  = lane & 15;
    const int koff = half * 2;        // K-offset within a 4-wide k-step

    const float* xg   = x + ((size_t)b * 256 + g * DIN_) * HW_;  // [k][p]
    const float* Wg   = W + (size_t)g * 160 * DIN_;              // [o][k]
    const float* bsg  = bias + g * 160;
    float* predb      = pred + (size_t)brel * J_ * I_ * D_;

    for (int nt = 0; nt < 3; ++nt) {
        const int p  = nt * 16 + l16;     // N index (column) for this lane
        const bool pv = (p < HW_);

        // B tiles (4x16): VGPR r, lanes<16 -> K=r, lanes>=16 -> K=r+2
        v2f b_lo, b_hi;
        b_lo[0] = pv ? xg[(0 + koff + 0) * HW_ + p] : 0.0f;
        b_lo[1] = pv ? xg[(0 + koff + 1) * HW_ + p] : 0.0f;
        b_hi[0] = pv ? xg[(4 + koff + 0) * HW_ + p] : 0.0f;
        b_hi[1] = pv ? xg[(4 + koff + 1) * HW_ + p] : 0.0f;

        for (int jt = 0; jt < J_; ++jt) {
            // A tiles (16x4): lane row = jt*16 + l16, K per layout above
            const float* arow = Wg + (jt * 16 + l16) * DIN_;
            v2f a_lo, a_hi;
            a_lo[0] = arow[0 + koff]; a_lo[1] = arow[0 + koff + 1];
            a_hi[0] = arow[4 + koff]; a_hi[1] = arow[4 + koff + 1];

            // C init with bias: C[r] covers (m = r + half*8) == d
            v8f c;
            #pragma unroll
            for (int r = 0; r < 8; ++r)
                c[r] = bsg[jt * 16 + half * 8 + r];

            c = __builtin_amdgcn_wmma_f32_16x16x4_f32(
                    false, a_lo, false, b_lo, (short)0, c, false, false);
            c = __builtin_amdgcn_wmma_f32_16x16x4_f32(
                    false, a_hi, false, b_hi, (short)0, c, false, false);

            if (pv) {
                const int i = g * HW_ + p;   // input-capsule index
                float* dst = predb + ((size_t)jt * I_ + i) * D_ + half * 8;
                float4 lo = make_float4(c[0], c[1], c[2], c[3]);
                float4 hi = make_float4(c[4], c[5], c[6], c[7]);
                *(float4*)(dst)     = lo;   // d = half*8 .. +3
                *(float4*)(dst + 4) = hi;   // d = half*8+4 .. +7
            }
        }
    }
}

// ---------------- Phase 2: dynamic routing, one block per batch element ----
#define RT_THREADS 256

__global__ __launch_bounds__(RT_THREADS) void cap_route(
    const float* __restrict__ pred,  // (chunk, J, I, D)
    float* __restrict__ out,         // (256, J, D)
    int b0)
{
    __shared__ float blog[J_][I_];    // routing logits, 46080 B
    __shared__ float invden[I_];      // softmax 1/denominator per i
    __shared__ float sacc[J_][D_];    // s accumulators
    __shared__ float vsh[J_][D_];     // squashed v

    const int tid  = threadIdx.x;
    const int brel = blockIdx.x;
    const float* predb = pred + (size_t)brel * J_ * I_ * D_;

    for (int idx = tid; idx < J_ * I_; idx += RT_THREADS)
        (&blog[0][0])[idx] = 0.0f;
    __syncthreads();

    for (int it = 0; it < 3; ++it) {
        // softmax denominators over j (per input capsule i)
        for (int i = tid; i < I_; i += RT_THREADS) {
            float s = 0.0f;
            #pragma unroll
            for (int j = 0; j < J_; ++j) s += __expf(blog[j][i]);
            invden[i] = 1.0f / s;
        }
        for (int idx = tid; idx < J_ * D_; idx += RT_THREADS)
            (&sacc[0][0])[idx] = 0.0f;
        __syncthreads();

        // s[j][d] = sum_i c[j][i] * pred[j][i][d]
        for (int j = 0; j < J_; ++j) {
            float acc[16];
            #pragma unroll
            for (int d = 0; d < 16; ++d) acc[d] = 0.0f;

            for (int i = tid; i < I_; i += RT_THREADS) {
                const float w = __expf(blog[j][i]) * invden[i];
                const float4* pr = (const float4*)(predb + ((size_t)j * I_ + i) * D_);
                float4 p0 = pr[0], p1 = pr[1], p2 = pr[2], p3 = pr[3];
                acc[0]  += w * p0.x; acc[1]  += w * p0.y; acc[2]  += w * p0.z; acc[3]  += w * p0.w;
                acc[4]  += w * p1.x; acc[5]  += w * p1.y; acc[6]  += w * p1.z; acc[7]  += w * p1.w;
                acc[8]  += w * p2.x; acc[9]  += w * p2.y; acc[10] += w * p2.z; acc[11] += w * p2.w;
                acc[12] += w * p3.x; acc[13] += w * p3.y; acc[14] += w * p3.z; acc[15] += w * p3.w;
            }
            // wave32 tree reduction, then one LDS atomic per wave
            #pragma unroll
            for (int d = 0; d < 16; ++d) {
                float v = acc[d];
                #pragma unroll
                for (int off = 16; off >= 1; off >>= 1)
                    v += __shfl_xor(v, off, 32);
                if ((tid & 31) == 0) atomicAdd(&sacc[j][d], v);
            }
        }
        __syncthreads();

        // squash: v = s * ||s|| / (1 + ||s||^2)
        if (tid < J_) {
            float n2 = 0.0f;
            #pragma unroll
            for (int d = 0; d < 16; ++d) { float s = sacc[tid][d]; n2 += s * s; }
            const float scale = sqrtf(n2) / (1.0f + n2);
            #pragma unroll
            for (int d = 0; d < 16; ++d) vsh[tid][d] = sacc[tid][d] * scale;
        }
        __syncthreads();

        // b[j][i] += dot(v[j], pred[j][i][:])
        for (int i = tid; i < I_; i += RT_THREADS) {
            #pragma unroll
            for (int j = 0; j < J_; ++j) {
                const float4* pr = (const float4*)(predb + ((size_t)j * I_ + i) * D_);
                float4 p0 = pr[0], p1 = pr[1], p2 = pr[2], p3 = pr[3];
                float dot =
                    vsh[j][0]*p0.x + vsh[j][1]*p0.y + vsh[j][2]*p0.z + vsh[j][3]*p0.w +
                    vsh[j][4]*p1.x + vsh[j][5]*p1.y + vsh[j][6]*p1.z + vsh[j][7]*p1.w +
                    vsh[j][8]*p2.x + vsh[j][9]*p2.y + vsh[j][10]*p2.z + vsh[j][11]*p2.w +
                    vsh[j][12]*p3.x + vsh[j][13]*p3.y + vsh[j][14]*p3.z + vsh[j][15]*p3.w;
                blog[j][i] += dot;
            }
        }
        __syncthreads();
    }

    // emit v for this batch element
    for (int idx = tid; idx < J_ * D_; idx += RT_THREADS)
        out[(size_t)(b0 + brel) * J_ * D_ + idx] = (&vsh[0][0])[idx];
}

extern "C" void kernel_launch(void* const* d_in, const int* in_sizes, int n_in,
                              void* d_out, int out_size, void* d_ws, size_t ws_size,
                              hipStream_t stream) {
    (void)in_sizes; (void)n_in; (void)out_size;
    const float* x    = (const float*)d_in[0];   // (256,256,6,6)
    const float* W    = (const float*)d_in[1];   // (5120,8)
    const float* bias = (const float*)d_in[2];   // (5120,)
    float* out  = (float*)d_out;                 // (256,10,16)
    float* pred = (float*)d_ws;                  // chunked (chunk, J, I, D)

    const size_t per_b = (size_t)J_ * I_ * D_ * sizeof(float);  // 737280 B
    int chunk = (int)(ws_size / per_b);
    if (chunk < 1)   chunk = 1;      // assumes ws_size >= ~720 KB
    if (chunk > BS_) chunk = BS_;

    for (int b0 = 0; b0 < BS_; b0 += chunk) {
        const int nb = (BS_ - b0 < chunk) ? (BS_ - b0) : chunk;
        cap_pred_wmma<<<nb * G_, 32, 0, stream>>>(x, W, bias, pred, b0);
        cap_route<<<nb, RT_THREADS, 0, stream>>>(pred, out, b0);
    }
}